// fMRIAlignment_9964324127027
// MI455X (gfx1250) — compile-verified
//
#include <hip/hip_runtime.h>
#include <hip/hip_bf16.h>

typedef __attribute__((ext_vector_type(2))) float v2f;
typedef __attribute__((ext_vector_type(8))) float v8f;

#define D_DIM 2048
#define O_DIM 8192
#define T_DIM 4096
#define NSLICE 16

// Stage 1: partial weighted row-sums of x.
// partial[slice][d] = sum_{w in slice, w>=1} hrf[D-w] * x[w][d]
__global__ void k_weighted_rows_partial(const float* __restrict__ x,
                                        const float* __restrict__ hrf,
                                        float* __restrict__ partial) {
    const int d = blockIdx.x * blockDim.x + threadIdx.x;       // column 0..D-1
    const int slice = blockIdx.y;                              // 0..NSLICE-1
    const int rowsPer = D_DIM / NSLICE;                        // 128
    int r0 = slice * rowsPer;
    const int r1 = r0 + rowsPer;
    if (r0 == 0) r0 = 1;                                       // loop starts at word=1
    float acc = 0.0f;
    for (int w = r0; w < r1; ++w) {
        // hrf[D-w] is wave-uniform -> scalar load; x access coalesced across d.
        acc = fmaf(hrf[D_DIM - w], x[(size_t)w * D_DIM + d], acc);
    }
    partial[(size_t)slice * D_DIM + d] = acc;
}

// Stage 2: deterministic reduction of the partials -> weighted_row[d]
__global__ void k_reduce_partials(const float* __restrict__ partial,
                                  float* __restrict__ wr) {
    const int d = blockIdx.x * blockDim.x + threadIdx.x;
    float acc = 0.0f;
    for (int s = 0; s < NSLICE; ++s)
        acc += partial[(size_t)s * D_DIM + d];
    wr[d] = acc;
}

// Stage 3: v = weighted_row @ weights via V_WMMA_F32_16X16X4_F32.
// A tile (16x4 f32): all 16 rows replicated with weighted_row[k0..k0+3],
// so every row of the 16x16 C tile equals the GEMV result for this 16-col
// slice. A-layout (ISA 7.12.2): lanes 0-15 hold K=0 (vgpr0) / K=1 (vgpr1),
// lanes 16-31 hold K=2 / K=3. B (4x16) mirrors that with N = lane&15.
// One wave per 16 output columns; K-loop over D in steps of 4.
__global__ void k_gemv_wmma(const float* __restrict__ wr,
                            const float* __restrict__ weights,
                            float* __restrict__ v) {
    const int lane = threadIdx.x & 31;
    const int wave = (blockIdx.x * blockDim.x + threadIdx.x) >> 5;
    const int n0   = wave * 16;                 // base output column
    const int ncol = n0 + (lane & 15);          // this lane's N
    const int kSh  = (lane >> 4) * 2;           // 0 for lanes 0-15, 2 for 16-31

    v8f c = {};
    const float* __restrict__ wcol = weights + ncol;
    for (int k0 = 0; k0 < D_DIM; k0 += 4) {
        v2f a, b;
        a.x = wr[k0 + kSh];
        a.y = wr[k0 + kSh + 1];
        b.x = wcol[(size_t)(k0 + kSh) * O_DIM];
        b.y = wcol[(size_t)(k0 + kSh + 1) * O_DIM];
        // 8 args: (neg_a, A, neg_b, B, c_mod, C, reuse_a, reuse_b)
        c = __builtin_amdgcn_wmma_f32_16x16x4_f32(false, a, false, b,
                                                  (short)0, c, false, false);
    }
    // Row M=0 of C lives in vgpr0 of lanes 0-15 (N = lane).
    if (lane < 16)
        v[n0 + lane] = c[0];
}

// Stage 4: out[t][o] = w_times_init[t][o] + v[o] + bias[t], 128-bit vectorized.
// This is the bandwidth-dominant stage (~256 MB of traffic).
__global__ void k_final_add(const float* __restrict__ w_init,
                            const float* __restrict__ v,
                            const float* __restrict__ bias,
                            float* __restrict__ out) {
    const size_t total4 = (size_t)T_DIM * O_DIM / 4;
    const size_t stride = (size_t)gridDim.x * blockDim.x;
    for (size_t i = (size_t)blockIdx.x * blockDim.x + threadIdx.x;
         i < total4; i += stride) {
        const size_t e = i * 4;
        const int t = (int)(e / O_DIM);
        const int o = (int)(e % O_DIM);
        const float4 wv = ((const float4*)w_init)[i];
        const float4 vv = *(const float4*)(v + o);
        const float  bb = bias[t];
        float4 r;
        r.x = wv.x + vv.x + bb;
        r.y = wv.y + vv.y + bb;
        r.z = wv.z + vv.z + bb;
        r.w = wv.w + vv.w + bb;
        ((float4*)out)[i] = r;
    }
}

extern "C" void kernel_launch(void* const* d_in, const int* in_sizes, int n_in,
                              void* d_out, int out_size, void* d_ws, size_t ws_size,
                              hipStream_t stream) {
    const float* x        = (const float*)d_in[0];   // [D, D]
    const float* hrf      = (const float*)d_in[1];   // [D]
    const float* weights  = (const float*)d_in[2];   // [D, O]
    const float* bias     = (const float*)d_in[3];   // [T, 1]
    const float* w_init   = (const float*)d_in[4];   // [T, O]
    float* out = (float*)d_out;                      // [T, O]

    float* ws      = (float*)d_ws;
    float* partial = ws;                              // NSLICE * D floats
    float* wr      = partial + (size_t)NSLICE * D_DIM; // D floats
    float* v       = wr + D_DIM;                      // O floats

    // Stage 1: x column-weighted row sums, sliced over rows for parallelism.
    dim3 g1(D_DIM / 256, NSLICE);
    k_weighted_rows_partial<<<g1, 256, 0, stream>>>(x, hrf, partial);

    // Stage 2: deterministic partial reduction.
    k_reduce_partials<<<D_DIM / 256, 256, 0, stream>>>(partial, wr);

    // Stage 3: GEMV via f32 WMMA. O/16 = 512 waves, 8 waves/block -> 64 blocks.
    k_gemv_wmma<<<(O_DIM / 16) / 8, 256, 0, stream>>>(wr, weights, v);

    // Stage 4: streaming broadcast-add, float4.
    k_final_add<<<4096, 256, 0, stream>>>(w_init, v, bias, out);
}